// MAPE_386547057267
// MI455X (gfx1250) — compile-verified
//
#include <hip/hip_runtime.h>

typedef float v2f __attribute__((ext_vector_type(2)));
typedef float v4f __attribute__((ext_vector_type(4)));
typedef float v8f __attribute__((ext_vector_type(8)));

constexpr int BLOCK = 256;       // 8 wave32 per block
constexpr int NW    = BLOCK / 32;

// Wave32 sum via V_WMMA_F32_16X16X4_F32.
// Each lane contributes {s0,s1} into the 16x4 f32 A operand (2 VGPRs/lane);
// B = ones(4x16) => D[m,n] = row-sum of A, identical in every column n.
// Summing the 8 C/D VGPRs gives each half-wave's column total; one shuffle
// folds the halves. Total = sum over all 32 lanes of (s0+s1), regardless of
// the exact lane->(m,k) mapping.
__device__ __forceinline__ float wave_sum_wmma(float s0, float s1) {
    v2f a; a.x = s0;   a.y = s1;
    v2f b; b.x = 1.0f; b.y = 1.0f;
    v8f c = {};
    c = __builtin_amdgcn_wmma_f32_16x16x4_f32(
            /*neg_a=*/false, a, /*neg_b=*/false, b,
            /*c_mod=*/(short)0, c, /*reuse_a=*/false, /*reuse_b=*/false);
    float w = ((c[0] + c[1]) + (c[2] + c[3])) + ((c[4] + c[5]) + (c[6] + c[7]));
    w += __shfl_down(w, 16, 32);   // lanes 0..15 now hold the full 32-lane sum
    return w;
}

__device__ __forceinline__ float ape(float p, float l) {
    return fabsf(p - l) * __builtin_amdgcn_rcpf(l);
}

__global__ void __launch_bounds__(BLOCK)
mape_partial_kernel(const float* __restrict__ pred, const float* __restrict__ lab,
                    float* __restrict__ ws, long long n) {
    __shared__ float red[NW];

    const long long nvec = n >> 2;                       // # of float4 elements
    const long long gid  = (long long)blockIdx.x * BLOCK + threadIdx.x;
    const long long nthr = (long long)gridDim.x * BLOCK;

    const v4f* __restrict__ p4 = (const v4f*)pred;
    const v4f* __restrict__ l4 = (const v4f*)lab;

    float s0 = 0.0f, s1 = 0.0f, s2 = 0.0f, s3 = 0.0f;

    // Each thread streams 2 consecutive float4s per array per iteration:
    // 4 x b128 NT loads in flight before the FMA chain consumes them.
    const long long stride2 = nthr * 2;
    for (long long i = gid * 2; i + 1 < nvec; i += stride2) {
        v4f pa = __builtin_nontemporal_load(&p4[i]);
        v4f la = __builtin_nontemporal_load(&l4[i]);
        v4f pb = __builtin_nontemporal_load(&p4[i + 1]);
        v4f lb = __builtin_nontemporal_load(&l4[i + 1]);
        s0 += ape(pa.x, la.x);
        s1 += ape(pa.y, la.y);
        s2 += ape(pa.z, la.z);
        s3 += ape(pa.w, la.w);
        s0 += ape(pb.x, lb.x);
        s1 += ape(pb.y, lb.y);
        s2 += ape(pb.z, lb.z);
        s3 += ape(pb.w, lb.w);
    }
    // Odd trailing float4 (if nvec is odd): one thread handles it.
    if ((nvec & 1) && gid == 0) {
        v4f pa = p4[nvec - 1];
        v4f la = l4[nvec - 1];
        s0 += ape(pa.x, la.x);
        s1 += ape(pa.y, la.y);
        s2 += ape(pa.z, la.z);
        s3 += ape(pa.w, la.w);
    }
    // Scalar tail (n % 4 != 0): at most 3 elements, low global threads.
    const long long tail = nvec << 2;
    if (gid < n - tail) {
        const long long i = tail + gid;
        s0 += ape(pred[i], lab[i]);
    }

    // Reconverged; EXEC all-1s -> WMMA legal.
    float w = wave_sum_wmma(s0 + s2, s1 + s3);

    const int lane = threadIdx.x & 31;
    const int wid  = threadIdx.x >> 5;
    if (lane == 0) red[wid] = w;
    __syncthreads();
    if (threadIdx.x == 0) {
        float b = 0.0f;
#pragma unroll
        for (int i = 0; i < NW; ++i) b += red[i];
        ws[blockIdx.x] = b;
    }
}

__global__ void __launch_bounds__(BLOCK)
mape_final_kernel(const float* __restrict__ ws, float* __restrict__ out,
                  int nparts, float scale) {
    __shared__ float red[NW];

    float s0 = 0.0f;
    for (int i = threadIdx.x; i < nparts; i += BLOCK) s0 += ws[i];

    float w = wave_sum_wmma(s0, 0.0f);

    const int lane = threadIdx.x & 31;
    const int wid  = threadIdx.x >> 5;
    if (lane == 0) red[wid] = w;
    __syncthreads();
    if (threadIdx.x == 0) {
        float b = 0.0f;
#pragma unroll
        for (int i = 0; i < NW; ++i) b += red[i];
        out[0] = b * scale;
    }
}

extern "C" void kernel_launch(void* const* d_in, const int* in_sizes, int n_in,
                              void* d_out, int out_size, void* d_ws, size_t ws_size,
                              hipStream_t stream) {
    const float* pred = (const float*)d_in[0];
    const float* lab  = (const float*)d_in[1];
    float*       out  = (float*)d_out;
    float*       ws   = (float*)d_ws;

    const long long n = (long long)in_sizes[0];

    // 2048 blocks x 256 threads (8 wave32 each): 512K threads, each streaming
    // 2 float4-pairs/iter over the 2^23-vector problem -> deep MLP for 23.3 TB/s.
    int blocks = 2048;
    const long long nvec = n >> 2;
    if (nvec < (long long)blocks * BLOCK * 2) {
        blocks = (int)((nvec + (BLOCK * 2 - 1)) / (BLOCK * 2));
        if (blocks < 1) blocks = 1;
    }
    if ((size_t)blocks * sizeof(float) > ws_size) blocks = (int)(ws_size / sizeof(float));
    if (blocks < 1) blocks = 1;

    mape_partial_kernel<<<blocks, BLOCK, 0, stream>>>(pred, lab, ws, n);

    const float scale = 100.0f / (float)n;
    mape_final_kernel<<<1, BLOCK, 0, stream>>>(ws, out, blocks, scale);
}